// dca_layer_74096775791084
// MI455X (gfx1250) — compile-verified
//
#include <hip/hip_runtime.h>
#include <hip/hip_bf16.h>
#include <math.h>

// Problem constants (from reference): x is (32, 512, 56, 56) f32.
#define B_   32
#define C_   512
#define HW_  3136            // 56*56
#define NPLANE (B_ * C_)     // 16384
#define KSZ  3
#define PADK 1

// mean_kernel tiling
#define PG      16           // planes per workgroup
#define TILE_K  64           // elements per plane per tile
#define NTILE   (HW_ / TILE_K)   // 49
#define RS      66           // LDS row stride in floats (64 + 2 pad -> bank spread)

typedef float v2f __attribute__((ext_vector_type(2)));
typedef float v4f __attribute__((ext_vector_type(4)));
typedef float v8f __attribute__((ext_vector_type(8)));

// ---------------------------------------------------------------------------
// Pass 1: y[b,c] = mean(x[b,c,:,:]).  Bandwidth bound (205 MB read).
// Reduction arithmetic runs on the matrix pipe: D = A(16x4 f32) * ones(4x16)
// accumulated in f32, so every D[m][n] ends up holding sum over plane m.
// ---------------------------------------------------------------------------
__global__ __launch_bounds__(256) void mean_kernel(const float* __restrict__ x,
                                                   float* __restrict__ y) {
  __shared__ float tile[PG * RS];
  __shared__ float partial[8][PG];

  const int tid  = threadIdx.x;
  const int lane = tid & 31;
  const int wave = tid >> 5;          // 0..7
  const int lp   = tid >> 4;          // loader: plane row 0..15
  const int lj   = tid & 15;          // loader: column group 0..15 (x4 floats)
  const long pg  = blockIdx.x;        // plane group, 0..1023
  const float* xg = x + pg * (long)PG * HW_;

  v8f acc = {};                       // 16 plane partial sums (all N columns equal)
  v2f bones; bones[0] = 1.0f; bones[1] = 1.0f;   // B = ones(4x16)

  const int am   = lane & 15;                 // A-matrix row M
  const int ksub = (lane < 16) ? 0 : 2;       // lanes 16-31 carry K=2,3
  const int kk0  = wave * 8;                  // this wave's K sub-range in the tile

  for (int t = 0; t < NTILE; ++t) {
    // Coalesced 16x64 tile load: 16 lanes cover one 256B plane row segment.
    const v4f v = *(const v4f*)(xg + (long)lp * HW_ + t * TILE_K + lj * 4);
    *(v4f*)&tile[lp * RS + lj * 4] = v;
    __syncthreads();

    const float* row = &tile[am * RS];
    v2f a0; a0[0] = row[kk0 + ksub];     a0[1] = row[kk0 + ksub + 1];
    acc = __builtin_amdgcn_wmma_f32_16x16x4_f32(false, a0, false, bones,
                                                (short)0, acc, false, false);
    v2f a1; a1[0] = row[kk0 + 4 + ksub]; a1[1] = row[kk0 + 4 + ksub + 1];
    acc = __builtin_amdgcn_wmma_f32_16x16x4_f32(false, a1, false, bones,
                                                (short)0, acc, false, false);
    __syncthreads();
  }

  // D layout: VGPR r holds M=r (lanes 0-15) and M=8+r (lanes 16-31); every N
  // column is identical, so lanes with N==0 publish the 16 plane sums.
  if ((lane & 15) == 0) {
    const int half = (lane >> 4) * 8;
    #pragma unroll
    for (int r = 0; r < 8; ++r) partial[wave][half + r] = acc[r];
  }
  __syncthreads();

  if (tid < PG) {
    float s = 0.0f;
    #pragma unroll
    for (int w = 0; w < 8; ++w) s += partial[w][tid];
    y[pg * PG + tid] = s * (1.0f / (float)HW_);
  }
}

// ---------------------------------------------------------------------------
// Pass 2: tiny deformable channel attention on y (16K outputs, pure VALU).
// ---------------------------------------------------------------------------
__global__ __launch_bounds__(C_) void attn_kernel(const float* __restrict__ y,
                                                  const float* __restrict__ w_off,
                                                  const float* __restrict__ w_def,
                                                  const float* __restrict__ b_def,
                                                  float* __restrict__ attn) {
  __shared__ float ys[C_];
  const int b = blockIdx.x;
  const int c = threadIdx.x;
  ys[c] = y[b * C_ + c];
  __syncthreads();

  const float ym1 = (c >= 1)      ? ys[c - 1] : 0.0f;
  const float y0  = ys[c];
  const float yp1 = (c < C_ - 1)  ? ys[c + 1] : 0.0f;

  float out = b_def[0];
  #pragma unroll
  for (int k = 0; k < KSZ; ++k) {
    // conv_offset_local (cross-correlation, zero pad): in[c-1+j], j=0..2
    const float off = w_off[k * KSZ + 0] * ym1
                    + w_off[k * KSZ + 1] * y0
                    + w_off[k * KSZ + 2] * yp1;
    const float pos  = (float)c + (float)(k - PADK) + off;
    const float p0   = floorf(pos);
    const float frac = pos - p0;
    const int   i0   = (int)p0;
    const int   i1   = i0 + 1;
    const int   c0   = i0 < 0 ? 0 : (i0 > C_ - 1 ? C_ - 1 : i0);
    const int   c1   = i1 < 0 ? 0 : (i1 > C_ - 1 ? C_ - 1 : i1);
    const float v0   = (i0 >= 0 && i0 < C_) ? ys[c0] : 0.0f;
    const float v1   = (i1 >= 0 && i1 < C_) ? ys[c1] : 0.0f;
    out += w_def[k] * (v0 * (1.0f - frac) + v1 * frac);
  }
  attn[b * C_ + c] = 1.0f / (1.0f + expf(-out));
}

// ---------------------------------------------------------------------------
// Pass 3: out = attn[plane] * x.  Planes walked in REVERSE so the tail of x
// (still resident in the 192MB L2 from pass 1) is re-read from L2, not HBM.
// Output is write-once -> nontemporal B128 stores keep it out of L2.
// ---------------------------------------------------------------------------
__global__ __launch_bounds__(256) void scale_kernel(const float* __restrict__ x,
                                                    const float* __restrict__ attn,
                                                    float* __restrict__ out) {
  const int plane = (NPLANE - 1) - (int)blockIdx.x;  // reverse order for L2 reuse
  const float a   = attn[plane];                     // block-uniform -> scalar load
  const long base4 = (long)plane * (HW_ / 4);
  const v4f* x4 = (const v4f*)x;
  v4f*       o4 = (v4f*)out;
  for (int i = threadIdx.x; i < HW_ / 4; i += 256) {
    v4f v = x4[base4 + i];
    v *= a;
    __builtin_nontemporal_store(v, &o4[base4 + i]);
  }
}

// ---------------------------------------------------------------------------
extern "C" void kernel_launch(void* const* d_in, const int* in_sizes, int n_in,
                              void* d_out, int out_size, void* d_ws, size_t ws_size,
                              hipStream_t stream) {
  const float* x     = (const float*)d_in[0];   // (32,512,56,56)
  const float* w_off = (const float*)d_in[1];   // (3,1,3)
  const float* w_def = (const float*)d_in[2];   // (3,)
  const float* b_def = (const float*)d_in[3];   // scalar
  float* out  = (float*)d_out;

  float* y    = (float*)d_ws;                   // 16384 floats
  float* attn = y + NPLANE;                     // 16384 floats

  mean_kernel <<<NPLANE / PG, 256, 0, stream>>>(x, y);
  attn_kernel <<<B_, C_, 0, stream>>>(y, w_off, w_def, b_def, attn);
  scale_kernel<<<NPLANE, 256, 0, stream>>>(x, attn, out);
}